// PCALFrames_32006096289978
// MI455X (gfx1250) — compile-verified
//
#include <hip/hip_runtime.h>
#include <hip/hip_bf16.h>
#include <math.h>

typedef float v2f __attribute__((ext_vector_type(2)));
typedef float v8f __attribute__((ext_vector_type(8)));

#define K_NBR 64
#define NODES_PER_WAVE 5          // 5 nodes * 3 dims = 15 of 16 WMMA rows/cols used
#define WAVES_PER_BLOCK 8
#define BLOCK_THREADS (WAVES_PER_BLOCK * 32)
#define NODES_PER_BLOCK (NODES_PER_WAVE * WAVES_PER_BLOCK)
#define NODE_STRIDE 200           // K_NBR*3 = 192 used, padded to dodge bank aliasing

__device__ __forceinline__ float dot3(const float a[3], const float b[3]) {
    return a[0]*b[0] + a[1]*b[1] + a[2]*b[2];
}
__device__ __forceinline__ void cross3(const float a[3], const float b[3], float o[3]) {
    o[0] = a[1]*b[2] - a[2]*b[1];
    o[1] = a[2]*b[0] - a[0]*b[2];
    o[2] = a[0]*b[1] - a[1]*b[0];
}

// Eigenvector of symmetric 3x3 for eigenvalue lam via null-space cross products.
__device__ __forceinline__ bool eigvec3(float a00, float a01, float a02,
                                        float a11, float a12, float a22,
                                        float lam, float v[3]) {
    float r0[3] = {a00 - lam, a01, a02};
    float r1[3] = {a01, a11 - lam, a12};
    float r2[3] = {a02, a12, a22 - lam};
    float c0[3], c1[3], c2[3];
    cross3(r0, r1, c0);
    cross3(r0, r2, c1);
    cross3(r1, r2, c2);
    float n0 = dot3(c0, c0), n1 = dot3(c1, c1), n2 = dot3(c2, c2);
    const float* best = c0; float nb = n0;
    if (n1 > nb) { best = c1; nb = n1; }
    if (n2 > nb) { best = c2; nb = n2; }
    if (nb < 1e-24f) return false;
    float inv = rsqrtf(nb);
    v[0] = best[0]*inv; v[1] = best[1]*inv; v[2] = best[2]*inv;
    return true;
}

__global__ __launch_bounds__(BLOCK_THREADS)
void PCALFrames_32006096289978_kernel(const float* __restrict__ pos,
                                      const int* __restrict__ edge_src,
                                      const int* __restrict__ num_neighbors,
                                      float* __restrict__ out,
                                      int n) {
    // Per-wave LDS: staged edge vectors (5 nodes x 64 edges x 3) + WMMA C spill.
    __shared__ float evLDS[WAVES_PER_BLOCK * NODES_PER_WAVE * NODE_STRIDE];
    __shared__ float cLDS[WAVES_PER_BLOCK * 16 * 17];

    const int tid  = threadIdx.x;
    const int w    = tid >> 5;
    const int lane = tid & 31;
    const int half = lane >> 4;   // 0: lanes 0-15 (K 0..1), 1: lanes 16-31 (K 2..3)
    const int lm   = lane & 15;
    const int nodeBase = ((int)blockIdx.x * WAVES_PER_BLOCK + w) * NODES_PER_WAVE;

    float* ev = &evLDS[w * NODES_PER_WAVE * NODE_STRIDE];
    float* cw = &cLDS[w * 16 * 17];

    // ---- Stage edge vectors: 320 edges per wave, 10 per lane (coalesced idx read,
    //      b96 gather of pos[src]; pos table is L2-resident). Padded edges have
    //      src == dst upstream so ev == 0 falls out naturally.
    #pragma unroll
    for (int i = 0; i < (NODES_PER_WAVE * K_NBR) / 32; ++i) {
        int e   = i * 32 + lane;       // 0..319
        int j   = e >> 6;              // node slot 0..4
        int k   = e & (K_NBR - 1);
        int gid = nodeBase + j;
        float ex = 0.f, ey = 0.f, ez = 0.f;
        if (gid < n) {
            __builtin_prefetch(edge_src + (size_t)gid * K_NBR + k + 32, 0, 0); // global_prefetch_b8
            int src  = edge_src[(size_t)gid * K_NBR + k];
            float px = pos[3*gid + 0], py = pos[3*gid + 1], pz = pos[3*gid + 2];
            ex = pos[3*src + 0] - px;
            ey = pos[3*src + 1] - py;
            ez = pos[3*src + 2] - pz;
        }
        float* dst = &ev[j * NODE_STRIDE + k * 3];
        dst[0] = ex; dst[1] = ey; dst[2] = ez;
    }
    __syncthreads();

    // ---- Accumulate cov = E^T E for 5 nodes at once with V_WMMA_F32_16X16X4_F32.
    // A(16x4): row m = 3j+r holds ev_j[k][r]; B(4x16) mirrors it, and since the f32
    // A/B lane layouts coincide (lanes 0-15: K=0/1, lanes 16-31: K=2/3), the SAME
    // registers feed both operands. D's diagonal 3x3 blocks are the covariances.
    //
    // Lane lm==15 (A row 15 / B col 15) only pollutes C row/col 15, which is never
    // read -> no masking needed: clamp it onto node 4 and load unconditionally.
    // This keeps EXEC untouched (no s_and_saveexec around ds_loads).
    const int jA = (lm < 15) ? (lm / 3) : 4;
    const int rA = (lm < 15) ? (lm % 3) : 0;
    // Per-lane base; chunk c reads immediate offsets 12c (K=2*half) and 12c+3 (K=2*half+1).
    const float* evA = &ev[jA * NODE_STRIDE + rA + 6 * half];

    // Stage all 32 A-operand scalars first so the ds_loads pipeline under one
    // s_wait_dscnt, then issue 16 back-to-back chained WMMAs (D->C accumulate).
    float areg[32];
    #pragma unroll
    for (int c = 0; c < K_NBR / 4; ++c) {
        areg[2*c + 0] = evA[12*c + 0];
        areg[2*c + 1] = evA[12*c + 3];
    }

    v8f acc = {};
    #pragma unroll
    for (int c = 0; c < K_NBR / 4; ++c) {
        v2f A; A.x = areg[2*c + 0]; A.y = areg[2*c + 1];
        acc = __builtin_amdgcn_wmma_f32_16x16x4_f32(
            /*neg_a=*/false, A, /*neg_b=*/false, A,
            /*c_mod=*/(short)0, acc, /*reuse_a=*/false, /*reuse_b=*/false);
    }

    // ---- Spill accumulator: VGPR v holds (M=v, N=lane) lanes 0-15 and (M=v+8) above.
    #pragma unroll
    for (int v = 0; v < 8; ++v)
        cw[(v + 8 * half) * 17 + lm] = acc[v];
    __syncthreads();

    // ---- Lanes 0..4: closed-form eigh of this wave's 3x3 covariances.
    if (lane < NODES_PER_WAVE) {
        int gid = nodeBase + lane;
        if (gid < n) {
            float* o = out + (size_t)gid * 9;
            if (num_neighbors[gid] <= 1) {
                #pragma unroll
                for (int t = 0; t < 9; ++t) o[t] = 0.f;
            } else {
                const int d = 3 * lane;
                float a00 = cw[(d+0)*17 + (d+0)];
                float a01 = cw[(d+0)*17 + (d+1)];
                float a02 = cw[(d+0)*17 + (d+2)];
                float a11 = cw[(d+1)*17 + (d+1)];
                float a12 = cw[(d+1)*17 + (d+2)];
                float a22 = cw[(d+2)*17 + (d+2)];

                float p1 = a01*a01 + a02*a02 + a12*a12;
                float q  = (a00 + a11 + a22) * (1.0f / 3.0f);
                float b00 = a00 - q, b11 = a11 - q, b22 = a22 - q;
                float p2 = b00*b00 + b11*b11 + b22*b22 + 2.f*p1;

                float e0[3], e1[3], e2[3];
                if (p2 < 1e-22f) {
                    // (near) multiple of identity -> identity frame
                    e0[0]=1.f; e0[1]=0.f; e0[2]=0.f;
                    e1[0]=0.f; e1[1]=1.f; e1[2]=0.f;
                    e2[0]=0.f; e2[1]=0.f; e2[2]=1.f;
                } else {
                    float p  = sqrtf(p2 * (1.0f / 6.0f));
                    float ip = 1.f / p;
                    float c00 = b00*ip, c01 = a01*ip, c02 = a02*ip;
                    float c11 = b11*ip, c12 = a12*ip, c22 = b22*ip;
                    float detB = c00*(c11*c22 - c12*c12)
                               - c01*(c01*c22 - c12*c02)
                               + c02*(c01*c12 - c11*c02);
                    float rr  = fminf(1.f, fmaxf(-1.f, 0.5f * detB));
                    float phi = acosf(rr) * (1.0f / 3.0f);
                    float lBig = q + 2.f*p*cosf(phi);
                    float lSml = q + 2.f*p*cosf(phi + 2.0943951023931953f);
                    // eigenvector of largest eigenvalue
                    if (!eigvec3(a00,a01,a02,a11,a12,a22, lBig, e2)) {
                        e2[0]=0.f; e2[1]=0.f; e2[2]=1.f;
                    }
                    // eigenvector of smallest eigenvalue
                    if (!eigvec3(a00,a01,a02,a11,a12,a22, lSml, e0)) {
                        float ax[3] = {0.f, 0.f, 0.f};
                        float x = fabsf(e2[0]), y = fabsf(e2[1]), z = fabsf(e2[2]);
                        if (x <= y && x <= z)      ax[0] = 1.f;
                        else if (y <= z)           ax[1] = 1.f;
                        else                       ax[2] = 1.f;
                        cross3(e2, ax, e0);
                        float inv = rsqrtf(fmaxf(dot3(e0, e0), 1e-30f));
                        e0[0]*=inv; e0[1]*=inv; e0[2]*=inv;
                    }
                    cross3(e2, e0, e1);   // middle eigenvector (sign arbitrary, as in eigh)
                    float inv1 = rsqrtf(fmaxf(dot3(e1, e1), 1e-30f));
                    e1[0]*=inv1; e1[1]*=inv1; e1[2]*=inv1;
                }
                // frames = vecs^T: rows in ascending-eigenvalue order
                o[0]=e0[0]; o[1]=e0[1]; o[2]=e0[2];
                o[3]=e1[0]; o[4]=e1[1]; o[5]=e1[2];
                o[6]=e2[0]; o[7]=e2[1]; o[8]=e2[2];
            }
        }
    }
}

extern "C" void kernel_launch(void* const* d_in, const int* in_sizes, int n_in,
                              void* d_out, int out_size, void* d_ws, size_t ws_size,
                              hipStream_t stream) {
    const float* pos           = (const float*)d_in[0];
    const int*   edge_src      = (const int*)d_in[1];
    // d_in[2] = edge_dst: unused (dst == edge_index / K by construction)
    const int*   num_neighbors = (const int*)d_in[3];
    float*       out           = (float*)d_out;

    int n = in_sizes[3];  // number of nodes
    int blocks = (n + NODES_PER_BLOCK - 1) / NODES_PER_BLOCK;
    PCALFrames_32006096289978_kernel<<<blocks, BLOCK_THREADS, 0, stream>>>(
        pos, edge_src, num_neighbors, out, n);
}